// StressGNN_8237747274157
// MI455X (gfx1250) — compile-verified
//
#include <hip/hip_runtime.h>

#define NN 25000
#define NE 100000

typedef float v2f __attribute__((ext_vector_type(2)));
typedef float v4f __attribute__((ext_vector_type(4)));
typedef float v8f __attribute__((ext_vector_type(8)));

__device__ __forceinline__ v8f wmma4(v2f a, v2f b, v8f c) {
    // V_WMMA_F32_16X16X4_F32: D = A(16x4) * B(4x16) + C, full fp32
    return __builtin_amdgcn_wmma_f32_16x16x4_f32(
        /*neg_a=*/false, a, /*neg_b=*/false, b,
        /*c_mod=*/(short)0, c, /*reuse_a=*/false, /*reuse_b=*/false);
}

__global__ __launch_bounds__(256) void gnn_zero_kernel(float* p, int n) {
    int i = blockIdx.x * blockDim.x + threadIdx.x;
    int s = gridDim.x * blockDim.x;
    for (; i < n; i += s) p[i] = 0.0f;
}

// Edge pass: per-edge weight matrix via WMMA on the edge MLP, fused with the
// per-edge matvec on the WMMA accumulator layout, then atomic scatter-add.
template<int IN>
__global__ __launch_bounds__(256) void gnn_edge_kernel(
    const float* __restrict__ x,          // [NN, IN] source features
    const long long* __restrict__ eidx,   // [2, NE] int64
    const float* __restrict__ eattr,      // [NE, 4]
    const float* __restrict__ w_mlp,      // [4, IN*32]
    const float* __restrict__ b_mlp,      // [IN*32]
    float* __restrict__ agg,              // [NN, 32] accumulators
    float* __restrict__ cnt,              // [NN]
    int nEdges, int addCnt)
{
    constexpr int WC = IN * 32;
    __shared__ float lds_w[4 * WC];
    __shared__ float lds_b[WC];
    __shared__ __align__(16) float lds_x[8][16][IN];  // per-wave gathered x[src]
    __shared__ int   lds_dst[8][16];

    const int tid = threadIdx.x;
    for (int i = tid; i < 4 * WC; i += 256) lds_w[i] = w_mlp[i];
    for (int i = tid; i < WC; i += 256)     lds_b[i] = b_mlp[i];
    __syncthreads();

    const int lane = tid & 31;
    const int wave = tid >> 5;
    const int ml = lane & 15;
    const int kh = lane >> 4;                 // which K half (0/1)
    const int nGroups = nEdges >> 4;          // NE % 16 == 0
    const float* wr0 = &lds_w[(kh ? 2 : 0) * WC];
    const float* wr1 = &lds_w[(kh ? 3 : 1) * WC];

    for (int g = blockIdx.x * 8 + wave; g < nGroups; g += gridDim.x * 8) {
        const int e0 = g << 4;
        const int eMine = e0 + ml;
        // A tile: edge_attr 16x4 in WMMA A layout (one b64 load per lane)
        v2f a = *(const v2f*)(eattr + eMine * 4 + kh * 2);
        // gather src features (b128 chunks) + dst index for the 16 edges
        const int srcI = (int)eidx[eMine];
        if (kh == 0) lds_dst[wave][ml] = (int)eidx[nEdges + eMine];
        const int iBase = kh * (IN / 2);
        const v4f* xs = (const v4f*)(x + (size_t)srcI * IN + iBase);
        v4f* xd = (v4f*)(&lds_x[wave][ml][iBase]);
        #pragma unroll
        for (int j = 0; j < IN / 8; ++j) xd[j] = xs[j];
        // (intra-wave LDS ops are in-order; no barrier needed)

        v8f c0 = {};  // msg cols 0..15, WMMA C layout (M=edge, N=col)
        v8f c1 = {};  // msg cols 16..31
        for (int i = 0; i < IN; ++i) {
            float xv[8];
            #pragma unroll
            for (int r = 0; r < 8; ++r) xv[r] = lds_x[wave][r + 8 * kh][i];
            #pragma unroll
            for (int oh = 0; oh < 2; ++oh) {
                const int cb = i * 32 + oh * 16;
                v2f b; b.x = wr0[cb + ml]; b.y = wr1[cb + ml];
                v8f z = {};
                v8f w = wmma4(a, b, z);          // w_e tile for 16 edges
                const float bb = lds_b[cb + ml];
                #pragma unroll
                for (int r = 0; r < 8; ++r) {
                    float wv = w[r] + bb;
                    wv = wv > 0.0f ? wv : 0.0f;   // ReLU on edge MLP
                    if (oh == 0) c0[r] += xv[r] * wv;
                    else         c1[r] += xv[r] * wv;
                }
            }
        }
        // scatter-add messages into dst nodes
        #pragma unroll
        for (int r = 0; r < 8; ++r) {
            const int d = lds_dst[wave][r + 8 * kh];
            atomicAdd(&agg[d * 32 + ml],      c0[r]);
            atomicAdd(&agg[d * 32 + 16 + ml], c1[r]);
        }
        if (addCnt && kh == 0) atomicAdd(&cnt[lds_dst[wave][ml]], 1.0f);
    }
}

// Node combine: h_out = relu(agg/max(cnt,1) + x @ root + bias), root GEMM on WMMA.
// A loads use clamped row indices (no exec-mask diamonds); stores predicated.
template<int IN>
__global__ __launch_bounds__(256) void gnn_node_kernel(
    const float* __restrict__ xin,   // [NN, IN]
    const float* __restrict__ root,  // [IN, 32]
    const float* __restrict__ bias,  // [32]
    const float* __restrict__ agg,   // [NN, 32]
    const float* __restrict__ cnt,   // [NN]
    float* __restrict__ hout,        // [NN, 32]
    int nNodes)
{
    const int tid = threadIdx.x;
    const int lane = tid & 31;
    const int wave = tid >> 5;
    const int ml = lane & 15;
    const int kh = lane >> 4;
    const int nTiles = (nNodes + 15) >> 4;

    // loop-invariant B tiles (root weights) in registers
    v2f b0[IN / 4], b1[IN / 4];
    #pragma unroll
    for (int k = 0; k < IN / 4; ++k) {
        const int kr = k * 4 + kh * 2;
        b0[k].x = root[kr * 32 + ml];
        b0[k].y = root[(kr + 1) * 32 + ml];
        b1[k].x = root[kr * 32 + 16 + ml];
        b1[k].y = root[(kr + 1) * 32 + 16 + ml];
    }

    for (int t = blockIdx.x * 8 + wave; t < nTiles; t += gridDim.x * 8) {
        const int m0 = t << 4;
        const int row = m0 + ml;
        const int rowC = row < nNodes ? row : nNodes - 1;   // clamp, no branch
        const float* arow = xin + (size_t)rowC * IN + kh * 2;
        v8f c0 = {}; v8f c1 = {};
        #pragma unroll
        for (int k = 0; k < IN / 4; ++k) {
            v2f a = *(const v2f*)(arow + k * 4);            // b64 load
            c0 = wmma4(a, b0[k], c0);
            c1 = wmma4(a, b1[k], c1);
        }
        const float bi0 = bias[ml], bi1 = bias[16 + ml];
        #pragma unroll
        for (int r = 0; r < 8; ++r) {
            const int rr = m0 + r + 8 * kh;
            if (rr < nNodes) {
                const float cn = cnt[rr];
                const float inv = 1.0f / (cn > 1.0f ? cn : 1.0f);
                float v0 = c0[r] + bi0 + agg[rr * 32 + ml] * inv;
                float v1 = c1[r] + bi1 + agg[rr * 32 + 16 + ml] * inv;
                hout[rr * 32 + ml]      = v0 > 0.0f ? v0 : 0.0f;
                hout[rr * 32 + 16 + ml] = v1 > 0.0f ? v1 : 0.0f;
            }
        }
    }
}

// Head: out = relu(h @ w1 + b1) @ w2 + b2, first GEMM on WMMA, second as
// a lane reduction (shfl_xor within half-wave column groups).
__global__ __launch_bounds__(256) void gnn_head_kernel(
    const float* __restrict__ h,    // [NN, 32]
    const float* __restrict__ w1,   // [32, 32]
    const float* __restrict__ b1,   // [32]
    const float* __restrict__ w2,   // [32]
    const float* __restrict__ b2,   // [1]
    float* __restrict__ out,        // [NN]
    int nNodes)
{
    const int tid = threadIdx.x;
    const int lane = tid & 31;
    const int wave = tid >> 5;
    const int ml = lane & 15;
    const int kh = lane >> 4;
    const int nTiles = (nNodes + 15) >> 4;

    v2f p0[8], p1[8];
    #pragma unroll
    for (int k = 0; k < 8; ++k) {
        const int kr = k * 4 + kh * 2;
        p0[k].x = w1[kr * 32 + ml];
        p0[k].y = w1[(kr + 1) * 32 + ml];
        p1[k].x = w1[kr * 32 + 16 + ml];
        p1[k].y = w1[(kr + 1) * 32 + 16 + ml];
    }
    const float bb0 = b1[ml], bb1 = b1[16 + ml];
    const float w2a = w2[ml], w2b = w2[16 + ml];
    const float bias2 = b2[0];

    for (int t = blockIdx.x * 8 + wave; t < nTiles; t += gridDim.x * 8) {
        const int m0 = t << 4;
        const int row = m0 + ml;
        const int rowC = row < nNodes ? row : nNodes - 1;
        const float* arow = h + (size_t)rowC * 32 + kh * 2;
        v8f c0 = {}; v8f c1 = {};
        #pragma unroll
        for (int k = 0; k < 8; ++k) {
            v2f a = *(const v2f*)(arow + k * 4);
            c0 = wmma4(a, p0[k], c0);
            c1 = wmma4(a, p1[k], c1);
        }
        #pragma unroll
        for (int r = 0; r < 8; ++r) {
            float v0 = c0[r] + bb0; v0 = v0 > 0.0f ? v0 : 0.0f;
            float v1 = c1[r] + bb1; v1 = v1 > 0.0f ? v1 : 0.0f;
            float s = v0 * w2a + v1 * w2b;
            s += __shfl_xor(s, 8, 16);
            s += __shfl_xor(s, 4, 16);
            s += __shfl_xor(s, 2, 16);
            s += __shfl_xor(s, 1, 16);
            const int rr = m0 + r + 8 * kh;
            if (ml == 0 && rr < nNodes) out[rr] = s + bias2;
        }
    }
}

extern "C" void kernel_launch(void* const* d_in, const int* in_sizes, int n_in,
                              void* d_out, int out_size, void* d_ws, size_t ws_size,
                              hipStream_t stream) {
    (void)in_sizes; (void)n_in; (void)out_size; (void)ws_size;
    const float*     x      = (const float*)d_in[0];
    const long long* eidx   = (const long long*)d_in[1];   // int64
    const float*     eattr  = (const float*)d_in[2];
    const float*     w_mlp1 = (const float*)d_in[3];
    const float*     b_mlp1 = (const float*)d_in[4];
    const float*     root1  = (const float*)d_in[5];
    const float*     bias1  = (const float*)d_in[6];
    const float*     w_mlp2 = (const float*)d_in[7];
    const float*     b_mlp2 = (const float*)d_in[8];
    const float*     root2  = (const float*)d_in[9];
    const float*     bias2  = (const float*)d_in[10];
    const float*     w_mlp3 = (const float*)d_in[11];
    const float*     b_mlp3 = (const float*)d_in[12];
    const float*     root3  = (const float*)d_in[13];
    const float*     bias3  = (const float*)d_in[14];
    const float*     w_out1 = (const float*)d_in[15];
    const float*     b_out1 = (const float*)d_in[16];
    const float*     w_out2 = (const float*)d_in[17];
    const float*     b_out2 = (const float*)d_in[18];
    float* out = (float*)d_out;

    float* h1  = (float*)d_ws;                 // [NN,32]
    float* h2  = h1 + (size_t)NN * 32;         // [NN,32]
    float* agg = h2 + (size_t)NN * 32;         // [NN,32]
    float* cnt = agg + (size_t)NN * 32;        // [NN] (contiguous after agg)

    const int edgeBlocks = 782;   // ceil(6250 groups / 8 waves)
    const int nodeBlocks = 196;   // ceil(1563 tiles / 8 waves)

    // Layer 1: IN=8 (x -> h1); also builds cnt (identical for all layers)
    gnn_zero_kernel<<<512, 256, 0, stream>>>(agg, NN * 33);
    gnn_edge_kernel<8><<<edgeBlocks, 256, 0, stream>>>(x, eidx, eattr, w_mlp1, b_mlp1, agg, cnt, NE, 1);
    gnn_node_kernel<8><<<nodeBlocks, 256, 0, stream>>>(x, root1, bias1, agg, cnt, h1, NN);

    // Layer 2: IN=32 (h1 -> h2); cnt reused
    gnn_zero_kernel<<<512, 256, 0, stream>>>(agg, NN * 32);
    gnn_edge_kernel<32><<<edgeBlocks, 256, 0, stream>>>(h1, eidx, eattr, w_mlp2, b_mlp2, agg, cnt, NE, 0);
    gnn_node_kernel<32><<<nodeBlocks, 256, 0, stream>>>(h1, root2, bias2, agg, cnt, h2, NN);

    // Layer 3: IN=32 (h2 -> h1); cnt reused
    gnn_zero_kernel<<<512, 256, 0, stream>>>(agg, NN * 32);
    gnn_edge_kernel<32><<<edgeBlocks, 256, 0, stream>>>(h2, eidx, eattr, w_mlp3, b_mlp3, agg, cnt, NE, 0);
    gnn_node_kernel<32><<<nodeBlocks, 256, 0, stream>>>(h2, root3, bias3, agg, cnt, h1, NN);

    // Output head (h1 -> out)
    gnn_head_kernel<<<nodeBlocks, 256, 0, stream>>>(h1, w_out1, b_out1, w_out2, b_out2, out, NN);
}